// RGCN_model_77506979823953
// MI455X (gfx1250) — compile-verified
//
#include <hip/hip_runtime.h>

// ---------------------------------------------------------------------------
// RGCN (2 layers, L=3 relations) for MI455X / gfx1250.
// GEMMs use V_WMMA_F32_16X16X4_F32 (exact f32, matches f32 reference).
// Edge aggregation is the bandwidth-bound part -> float4 coalesced scatter
// with global_atomic_add_f32; self-loop pass is non-atomic (kernel-ordered).
// ---------------------------------------------------------------------------

typedef float v2f __attribute__((ext_vector_type(2)));
typedef float v8f __attribute__((ext_vector_type(8)));

#define N_NODES 100000
#define N_EDGES 1600000
#define NREL    3
#define IN_C    128
#define HID_C   64
#define OUT_C   16

// ---------------- degree / normalization ----------------

__global__ void init_deg_kernel(float* __restrict__ deg, int n) {
    int i = blockIdx.x * blockDim.x + threadIdx.x;
    if (i < n) deg[i] = 1.0f;   // self-loop contributes 1
}

__global__ void count_deg_kernel(const int* __restrict__ ei,  // [NREL,2,E]
                                 float* __restrict__ deg,     // [NREL,N]
                                 int E, int N) {
    long long t = (long long)blockIdx.x * blockDim.x + threadIdx.x;
    if (t >= (long long)NREL * E) return;
    int r = (int)(t / E);
    int e = (int)(t % E);
    int d = ei[(size_t)r * 2 * E + E + e];   // dst of relation r
    atomicAdd(&deg[(size_t)r * N + d], 1.0f);
}

__global__ void deg_to_dinv_kernel(float* __restrict__ deg, int n) {
    int i = blockIdx.x * blockDim.x + threadIdx.x;
    if (i < n) deg[i] = rsqrtf(deg[i]);      // deg >= 1 always (self-loop)
}

__global__ void zero_kernel(float* __restrict__ p, int n) {
    int i = blockIdx.x * blockDim.x + threadIdx.x;
    if (i < n) p[i] = 0.0f;
}

// ---------------- WMMA f32 GEMM:  C[N x Cc] = A[N x K] @ B[K x Cc] ----------
// One wave (32 lanes) computes one 16x16 output tile, looping K in steps of 4
// with V_WMMA_F32_16X16X4_F32.  N % 16 == 0, K % 4 == 0, Cc % 16 == 0.
//
// Fragment layouts (CDNA5 ISA 7.12.2):
//  A 16x4 f32 : lane L holds row M = L%16, K = 2*(L/16) + {0,1} in a.{x,y}
//  B 4x16 f32 : lane L holds col N = L%16, K = 2*(L/16) + {0,1} in b.{x,y}
//  C 16x16    : lane L, vgpr v  ->  row M = v + 8*(L/16), col N = L%16

__global__ __launch_bounds__(32)
void wmma_gemm_f32_kernel(const float* __restrict__ A,
                          const float* __restrict__ B,
                          float* __restrict__ C,
                          int K, int Cc) {
    const int lane = threadIdx.x;        // 0..31
    const int half = lane >> 4;          // 0 / 1
    const int lid  = lane & 15;
    const int row  = blockIdx.x * 16 + lid;   // A row for this lane's frags
    const int col  = blockIdx.y * 16 + lid;   // B/C col for this lane

    const float* __restrict__ arow = A + (size_t)row * K;
    v8f acc = {};

    for (int kk = 0; kk < K; kk += 4) {
        const int ka = kk + 2 * half;
        v2f a, b;
        a.x = arow[ka];
        a.y = arow[ka + 1];
        b.x = B[(size_t)ka * Cc + col];
        b.y = B[(size_t)(ka + 1) * Cc + col];
        acc = __builtin_amdgcn_wmma_f32_16x16x4_f32(
            /*neg_a=*/false, a, /*neg_b=*/false, b,
            /*c_mod=*/(short)0, acc, /*reuse_a=*/false, /*reuse_b=*/false);
    }

#pragma unroll
    for (int v = 0; v < 8; ++v) {
        const int r = blockIdx.x * 16 + v + 8 * half;
        C[(size_t)r * Cc + col] = acc[v];
    }
}

// ---------------- edge scatter (atomic) ----------------
// C/4 threads per edge; each thread moves one float4 of channels.
// out[dst] += H[src] * dinv[src] * dinv[dst]

template <int C>
__global__ void scatter_edges_kernel(const float* __restrict__ H,
                                     const float* __restrict__ dinv,
                                     const int* __restrict__ src,
                                     const int* __restrict__ dst,
                                     float* __restrict__ out, int E) {
    constexpr int CPG = C / 4;
    long long t = (long long)blockIdx.x * blockDim.x + threadIdx.x;
    long long e = t / CPG;
    if (e >= E) return;
    const int c4 = (int)(t % CPG) * 4;

    const int s = src[e];
    const int d = dst[e];
    const float norm = dinv[s] * dinv[d];

    const float4 v = *(const float4*)(H + (size_t)s * C + c4);
    float* op = out + (size_t)d * C + c4;
    atomicAdd(op + 0, v.x * norm);
    atomicAdd(op + 1, v.y * norm);
    atomicAdd(op + 2, v.z * norm);
    atomicAdd(op + 3, v.w * norm);
}

// ---------------- self-loop add (non-atomic; runs after scatter) -----------

__global__ void self_loop_kernel(const float* __restrict__ H,
                                 const float* __restrict__ dinv,
                                 float* __restrict__ out, int N, int C) {
    int i = blockIdx.x * blockDim.x + threadIdx.x;
    if (i >= N * C) return;
    const int n = i / C;
    const float w = dinv[n] * dinv[n];
    out[i] += H[i] * w;
}

// ---------------- epilogues ----------------

__global__ void bias_relu_kernel(float* __restrict__ hid,
                                 const float* __restrict__ b1, int N) {
    int i = blockIdx.x * blockDim.x + threadIdx.x;
    if (i >= N * HID_C) return;
    const int c = i & (HID_C - 1);
    const float b = b1[c] + b1[HID_C + c] + b1[2 * HID_C + c];
    const float v = hid[i] + b;
    hid[i] = v > 0.0f ? v : 0.0f;
}

__global__ void bias_add_kernel(float* __restrict__ out,
                                const float* __restrict__ b2, int N) {
    int i = blockIdx.x * blockDim.x + threadIdx.x;
    if (i >= N * OUT_C) return;
    const int c = i & (OUT_C - 1);
    out[i] += b2[c] + b2[OUT_C + c] + b2[2 * OUT_C + c];
}

// ---------------------------------------------------------------------------

extern "C" void kernel_launch(void* const* d_in, const int* in_sizes, int n_in,
                              void* d_out, int out_size, void* d_ws, size_t ws_size,
                              hipStream_t stream) {
    const float* x  = (const float*)d_in[0];   // [N, 128]
    const float* W1 = (const float*)d_in[1];   // [3, 128, 64]
    const float* b1 = (const float*)d_in[2];   // [3, 64]
    const float* W2 = (const float*)d_in[3];   // [3, 64, 16]
    const float* b2 = (const float*)d_in[4];   // [3, 16]
    const int*   ei = (const int*)d_in[5];     // [3, 2, E]
    float* out = (float*)d_out;                // [N, 16]

    const int N = N_NODES, E = N_EDGES;

    // workspace layout (floats): dinv[3*N] | hid[N*64] | hbuf[N*64]
    float* dinv = (float*)d_ws;
    float* hid  = dinv + (size_t)NREL * N;
    float* hbuf = hid + (size_t)N * HID_C;

    const int TB = 256;
    auto nb = [](long long n, int tb) { return (unsigned)((n + tb - 1) / tb); };

    // ---- degree -> dinv (shared by both layers) ----
    init_deg_kernel<<<nb((long long)NREL * N, TB), TB, 0, stream>>>(dinv, NREL * N);
    count_deg_kernel<<<nb((long long)NREL * E, TB), TB, 0, stream>>>(ei, dinv, E, N);
    deg_to_dinv_kernel<<<nb((long long)NREL * N, TB), TB, 0, stream>>>(dinv, NREL * N);

    // ---- layer 1: hid = sum_r scatter_r(x @ W1[r]) ----
    zero_kernel<<<nb((long long)N * HID_C, TB), TB, 0, stream>>>(hid, N * HID_C);
    for (int r = 0; r < NREL; ++r) {
        const int* src = ei + (size_t)r * 2 * E;
        const int* dst = src + E;
        const float* dv = dinv + (size_t)r * N;

        wmma_gemm_f32_kernel<<<dim3(N / 16, HID_C / 16), 32, 0, stream>>>(
            x, W1 + (size_t)r * IN_C * HID_C, hbuf, IN_C, HID_C);

        scatter_edges_kernel<HID_C>
            <<<nb((long long)E * (HID_C / 4), TB), TB, 0, stream>>>(
                hbuf, dv, src, dst, hid, E);

        self_loop_kernel<<<nb((long long)N * HID_C, TB), TB, 0, stream>>>(
            hbuf, dv, hid, N, HID_C);
    }
    bias_relu_kernel<<<nb((long long)N * HID_C, TB), TB, 0, stream>>>(hid, b1, N);

    // ---- layer 2: out = sum_r scatter_r(hid @ W2[r]) ----
    zero_kernel<<<nb((long long)N * OUT_C, TB), TB, 0, stream>>>(out, N * OUT_C);
    for (int r = 0; r < NREL; ++r) {
        const int* src = ei + (size_t)r * 2 * E;
        const int* dst = src + E;
        const float* dv = dinv + (size_t)r * N;

        wmma_gemm_f32_kernel<<<dim3(N / 16, OUT_C / 16), 32, 0, stream>>>(
            hid, W2 + (size_t)r * HID_C * OUT_C, hbuf, HID_C, OUT_C);

        scatter_edges_kernel<OUT_C>
            <<<nb((long long)E * (OUT_C / 4), TB), TB, 0, stream>>>(
                hbuf, dv, src, dst, out, E);

        self_loop_kernel<<<nb((long long)N * OUT_C, TB), TB, 0, stream>>>(
            hbuf, dv, out, N, OUT_C);
    }
    bias_add_kernel<<<nb((long long)N * OUT_C, TB), TB, 0, stream>>>(out, b2, N);
}